// APPNPConv_9174050144817
// MI455X (gfx1250) — compile-verified
//
#include <hip/hip_runtime.h>

#define ALPHA 0.2f
#define D 64
#define WAVES_PER_BLOCK 4   // 128 threads; 8KB LDS gather buffer per wave -> 32KB/block

typedef __attribute__((ext_vector_type(4))) int v4i;

// ---- CDNA5 async global->LDS gather (ASYNCcnt path) ------------------------
// Probe-revealed signature: param0 is int4* in AS1 (__device__), non-const.
__device__ __forceinline__ void async_gather_b128(const float* gsrc, float* ldst) {
#if __has_builtin(__builtin_amdgcn_global_load_async_to_lds_b128)
    __builtin_amdgcn_global_load_async_to_lds_b128(
        (__attribute__((address_space(1))) v4i*)gsrc,
        (__attribute__((address_space(3))) v4i*)ldst,
        0, 0);
#else
    // VDST = LDS byte address VGPR, VADDR = 64-bit global address (GV mode).
    unsigned laddr = (unsigned)(unsigned long long)
        (__attribute__((address_space(3))) void*)ldst;
    asm volatile("global_load_async_to_lds_b128 %0, %1, off"
                 :: "v"(laddr), "v"(gsrc)
                 : "memory");
#endif
}

__device__ __forceinline__ void wait_async0() {
#if __has_builtin(__builtin_amdgcn_s_wait_asynccnt)
    __builtin_amdgcn_s_wait_asynccnt(0);
#else
    asm volatile("s_wait_asynccnt 0x0" ::: "memory");
#endif
}
// ----------------------------------------------------------------------------

__device__ __forceinline__ int lower_bound_i32(const int* __restrict__ arr, int n, int key) {
    int lo = 0, hi = n;
    while (lo < hi) {
        int mid = (lo + hi) >> 1;
        if (arr[mid] < key) lo = mid + 1; else hi = mid;
    }
    return lo;
}

// One wave32 per node.
//  1) lanes 0/1 binary-search the sorted edge_row for this node's segment
//  2) 32 edges staged cooperatively (coalesced col/val loads), broadcast via
//     wave32 shuffles (ds_bpermute)
//  3) all gathered h-rows of the batch launched as fire-and-forget
//     GLOBAL_LOAD_ASYNC_TO_LDS_B128 (16-deep MLP, no landing VGPRs)
//  4) one s_wait_asynccnt 0, then rows streamed from LDS (ds_load_b128) + FMA
//  5) cross-half shuffle reduce, axpy with h0, float4 store
__global__ __launch_bounds__(32 * WAVES_PER_BLOCK) void appnp_spmm_kernel(
    const int*   __restrict__ edge_row,
    const int*   __restrict__ edge_col,
    const float* __restrict__ edge_vals,
    const float* __restrict__ h,
    const float* __restrict__ h0,
    float*       __restrict__ out,
    int n_nodes, int n_edges)
{
    __shared__ float lbuf[WAVES_PER_BLOCK][32 * D];   // 8KB gather buffer per wave

    const int lane = threadIdx.x & 31;
    const int wave = threadIdx.x >> 5;
    const int node = blockIdx.x * WAVES_PER_BLOCK + wave;
    if (node >= n_nodes) return;

    float* mybuf = &lbuf[wave][0];

    // Segment bounds in the sorted row array.
    int b = 0;
    if (lane < 2) b = lower_bound_i32(edge_row, n_edges, node + lane);
    const int seg_start = __shfl(b, 0, 32);
    const int seg_end   = __shfl(b, 1, 32);

    const int half  = lane >> 4;   // which edge of the pair this half-wave handles
    const int qlane = lane & 15;   // 16 lanes x float4 = 64 features

    float4 acc; acc.x = 0.f; acc.y = 0.f; acc.z = 0.f; acc.w = 0.f;

    for (int base = seg_start; base < seg_end; base += 32) {
        // Coalesced staging of up to 32 (col, val) pairs into registers.
        const int e = base + lane;
        int   c = 0;
        float v = 0.f;
        if (e < seg_end) {
            c = edge_col[e];
            v = edge_vals[e];
        }
        const int cnt = min(32, seg_end - base);

        // Issue: each instruction moves 2 rows (16B/lane x 32 lanes = 512B).
        // Identical index structure to the consume loop below, so every row
        // that is consumed has been written (padded lanes carry c == 0, a
        // valid finite row, and v == 0 so they contribute exactly zero).
        for (int j = 0; j < cnt; j += 2) {
            const int kk = (j + half) & 31;
            const int ck = __shfl(c, kk, 32);
            async_gather_b128(h + (size_t)ck * D + qlane * 4,
                              mybuf + kk * D + qlane * 4);
        }

        wait_async0();   // all batch gathers resident in LDS

        // Consume: stream rows from LDS, FMA with broadcast edge values.
        for (int k = 0; k < cnt; k += 2) {
            const int   kk = (k + half) & 31;
            const float vk = __shfl(v, kk, 32);
            const float4 hr = *(const float4*)(mybuf + kk * D + qlane * 4);
            acc.x += vk * hr.x;
            acc.y += vk * hr.y;
            acc.z += vk * hr.z;
            acc.w += vk * hr.w;
        }
        // Register deps force DScnt completion before the buffer is reused.
    }

    // Fold half-wave 1's partial sums into lanes 0..15.
    acc.x += __shfl(acc.x, (lane + 16) & 31, 32);
    acc.y += __shfl(acc.y, (lane + 16) & 31, 32);
    acc.z += __shfl(acc.z, (lane + 16) & 31, 32);
    acc.w += __shfl(acc.w, (lane + 16) & 31, 32);

    if (lane < 16) {
        const float4 r = *(const float4*)(h0 + (size_t)node * D + qlane * 4);
        float4 o;
        o.x = (1.0f - ALPHA) * acc.x + ALPHA * r.x;
        o.y = (1.0f - ALPHA) * acc.y + ALPHA * r.y;
        o.z = (1.0f - ALPHA) * acc.z + ALPHA * r.z;
        o.w = (1.0f - ALPHA) * acc.w + ALPHA * r.w;
        *(float4*)(out + (size_t)node * D + qlane * 4) = o;
    }
}

extern "C" void kernel_launch(void* const* d_in, const int* in_sizes, int n_in,
                              void* d_out, int out_size, void* d_ws, size_t ws_size,
                              hipStream_t stream) {
    const int*   edge_row  = (const int*)  d_in[0];
    const int*   edge_col  = (const int*)  d_in[1];
    const float* edge_vals = (const float*)d_in[2];
    const float* h         = (const float*)d_in[3];
    const float* h0        = (const float*)d_in[4];
    float*       out       = (float*)d_out;

    const int n_edges = in_sizes[0];
    const int n_nodes = out_size / D;

    dim3 block(32 * WAVES_PER_BLOCK);
    dim3 grid((n_nodes + WAVES_PER_BLOCK - 1) / WAVES_PER_BLOCK);
    appnp_spmm_kernel<<<grid, block, 0, stream>>>(edge_row, edge_col, edge_vals,
                                                  h, h0, out, n_nodes, n_edges);
}